// Net_7842610283237
// MI455X (gfx1250) — compile-verified
//
#include <hip/hip_runtime.h>
#include <hip/hip_bf16.h>

typedef __attribute__((ext_vector_type(2))) float v2f;
typedef __attribute__((ext_vector_type(8))) float v8f;
typedef __attribute__((ext_vector_type(4))) unsigned int v4u;
typedef __attribute__((ext_vector_type(8))) int v8i;
typedef __attribute__((ext_vector_type(4))) int v4i;

#define NNODES 4096

// ---------------- degree -> D^{-1/2} ----------------
__global__ void deg_rsqrt_kernel(const float* __restrict__ adj, float* __restrict__ dinv) {
    __shared__ float s[256];
    int row = blockIdx.x;
    int tid = threadIdx.x;
    const float* rp = adj + (size_t)row * NNODES;
    float sum = 0.f;
    for (int j = tid; j < NNODES; j += 256) sum += rp[j];
    s[tid] = sum;
    __syncthreads();
    for (int off = 128; off > 0; off >>= 1) {
        if (tid < off) s[tid] += s[tid + off];
        __syncthreads();
    }
    if (tid == 0) dinv[row] = rsqrtf(fmaxf(s[0], 1e-12f));
}

// ---------------- L = I - D^{-1/2} A D^{-1/2} ----------------
__global__ void laplacian_kernel(const float* __restrict__ adj, const float* __restrict__ dinv,
                                 float* __restrict__ L) {
    int j = blockIdx.x * 256 + threadIdx.x;
    int i = blockIdx.y;
    float v = -dinv[i] * adj[(size_t)i * NNODES + j] * dinv[j];
    if (i == j) v += 1.f;
    L[(size_t)i * NNODES + j] = v;
}

// ---------------- zero-padded pack: src[rows,cols] -> dst[rowsP,colsP] ----------------
__global__ void pack_pad_kernel(const float* __restrict__ src, float* __restrict__ dst,
                                int rows, int cols, int colsP, int total) {
    int idx = blockIdx.x * 256 + threadIdx.x;
    if (idx >= total) return;
    int r = idx / colsP, c = idx - r * colsP;
    dst[idx] = (r < rows && c < cols) ? src[r * cols + c] : 0.f;
}

// ---------------- final copy [4096,16pad] -> [4096,4] ----------------
__global__ void copy_out_kernel(const float* __restrict__ src, float* __restrict__ dst) {
    int idx = blockIdx.x * 256 + threadIdx.x;
    if (idx >= NNODES * 4) return;
    int r = idx >> 2, c = idx & 3;
    dst[idx] = src[r * 16 + c];
}

// ---------------- TDM: 2-D tile DMA global -> LDS (D# per CDNA5 ISA 8.3/8.4) ----------
// dim0 contiguous (elements), dim1 rows with stride0 (elements). f32 data (data_size=2).
// padInterval/padAmount implement the banked LDS row stride.
__device__ __forceinline__ void tdm_load_2d(unsigned ldsByteOff, const float* gptr,
                                            unsigned dim0, unsigned dim1,
                                            unsigned tile0, unsigned tile1,
                                            unsigned long long stride0,
                                            unsigned padInterval, unsigned padAmount) {
    unsigned long long ga = (unsigned long long)(uintptr_t)gptr;
    v4u g0;
    g0[0] = 1u;                                   // count=1, user descriptor
    g0[1] = ldsByteOff;                           // lds_addr
    g0[2] = (unsigned)ga;                         // global_addr[31:0]
    g0[3] = (unsigned)((ga >> 32) & 0x1FFFFFFull) // global_addr[56:32]
            | (2u << 30);                         // type=2 ("image")
    v8i g1;
    g1[0] = (int)((2u << 16)                      // data_size = 4 bytes
                  | (1u << 20)                    // pad_enable
                  | (padInterval << 22)           // pad every 2^(v+1) dwords
                  | (padAmount << 25));           // pad of (v+1) dwords
    g1[1] = (int)((dim0 & 0xFFFFu) << 16);        // tensor_dim0[15:0]
    g1[2] = (int)((dim0 >> 16) | ((dim1 & 0xFFFFu) << 16)); // dim0[31:16] | dim1[15:0]
    g1[3] = (int)((dim1 >> 16) | (tile0 << 16));  // dim1[31:16] | tile_dim0
    g1[4] = (int)(tile1 & 0xFFFFu);               // tile_dim1 (tile_dim2=0 unused)
    g1[5] = (int)(unsigned)(stride0 & 0xFFFFFFFFull);       // dim0_stride[31:0]
    g1[6] = (int)(unsigned)((stride0 >> 32) & 0xFFFFull);   // dim0_stride[47:32]
    g1[7] = 0;
    v4i gz4 = {0, 0, 0, 0};
    v8i gz8 = {0, 0, 0, 0, 0, 0, 0, 0};
    __builtin_amdgcn_tensor_load_to_lds(g0, g1, gz4, gz4, gz8, 0);
}

// ---------------- fp32 WMMA GEMM: D = alpha*(A@B) + beta*C (+bias)(+relu) ----------------
// M = 4096 (gridDim.y*128), K mult of 16, N padded mult of 16 (guarded to Np).
// Block tile 128x64, 8 waves (2x4); wave tile 64x16 = 4 stacked 16x16 WMMA tiles.
// Tiles staged by the Tensor Data Mover (wave 0 issues D#), double-buffered in LDS:
//   A: 128x16 padded to row stride 20 floats  (buf b at byte 10240*b)
//   B: 16x64  padded to row stride 80 floats  (buf b at byte 20480 + 5120*b)
__launch_bounds__(256)
__global__ void wmma_gemm_f32(const float* __restrict__ A, const float* __restrict__ B,
                              const float* __restrict__ Cin, float* __restrict__ D,
                              const float* __restrict__ bias,
                              int K, int Np, int lda, int ldb, int ldc, int ldd,
                              float alpha, float beta, int relu) {
    extern __shared__ float smem[]; // dynamic LDS starts at offset 0 (no static shared)

    int tid    = threadIdx.x;
    int lane   = tid & 31;
    int laneLo = lane & 15;
    int laneHi = lane >> 4;        // 0: K pair {0,1}, 1: K pair {2,3}
    int wave   = tid >> 5;
    int m0     = (wave >> 2) * 64; // wave row base within block tile
    int n0     = (wave & 3) * 16;  // wave col base within block tile
    int blockM = blockIdx.y * 128;
    int blockN = blockIdx.x * 64;

    v8f acc[4] = {};

    int nk = K >> 4;

    // preload slab 0 into buffer 0
    if (wave == 0) {
        tdm_load_2d(0u, A + (size_t)blockM * lda, (unsigned)K, 128u, 16u, 128u,
                    (unsigned long long)lda, 3u, 3u);            // 16dw rows + 4dw pad
        tdm_load_2d(20480u, B + blockN, (unsigned)(Np - blockN), (unsigned)K, 64u, 16u,
                    (unsigned long long)ldb, 5u, 15u);           // 64dw rows + 16dw pad
    }

    for (int ik = 0; ik < nk; ++ik) {
        int kk  = ik << 4;
        int cur = ik & 1;
        if (wave == 0) {
            if (ik + 1 < nk) {
                int nxt = cur ^ 1;
                int kn  = kk + 16;
                tdm_load_2d((unsigned)(nxt * 10240),
                            A + (size_t)blockM * lda + kn, (unsigned)(K - kn), 128u,
                            16u, 128u, (unsigned long long)lda, 3u, 3u);
                tdm_load_2d((unsigned)(20480 + nxt * 5120),
                            B + (size_t)kn * ldb + blockN, (unsigned)(Np - blockN),
                            (unsigned)(K - kn), 64u, 16u, (unsigned long long)ldb, 5u, 15u);
                __builtin_amdgcn_s_wait_tensorcnt((short)2); // current bufs (oldest 2) done
            } else {
                __builtin_amdgcn_s_wait_tensorcnt((short)0); // final slab done
            }
        }
        __syncthreads();

        const float* sA = smem + cur * 2560;
        const float* sB = smem + 5120 + cur * 1280;

        #pragma unroll
        for (int k4 = 0; k4 < 4; ++k4) {
            int kb = k4 * 4 + laneHi * 2;
            // B frag 4x16: VGPR0 = rows {0,2}, VGPR1 = rows {1,3} across lane halves
            v2f bf;
            bf.x = sB[(kb + 0) * 80 + n0 + laneLo];
            bf.y = sB[(kb + 1) * 80 + n0 + laneLo];
            #pragma unroll
            for (int mt = 0; mt < 4; ++mt) {
                int ar = (m0 + mt * 16 + laneLo) * 20 + kb;
                v2f af; // A frag 16x4: lanes 0-15 K={0,1}, lanes 16-31 K={2,3}
                af.x = sA[ar];
                af.y = sA[ar + 1];
                acc[mt] = __builtin_amdgcn_wmma_f32_16x16x4_f32(
                    false, af, false, bf, (short)0, acc[mt], false, false);
            }
        }
        __syncthreads(); // all waves done with 'cur' before TDM overwrites it
    }

    // epilogue: C/D layout -> VGPR r: lanes 0-15 row r, lanes 16-31 row r+8
    #pragma unroll
    for (int mt = 0; mt < 4; ++mt) {
        #pragma unroll
        for (int r = 0; r < 8; ++r) {
            int row = blockM + m0 + mt * 16 + laneHi * 8 + r;
            int col = blockN + n0 + laneLo;
            if (col < Np) {
                float v = alpha * acc[mt][r];
                if (beta != 0.f) v += beta * Cin[(size_t)row * ldc + col];
                if (bias) v += bias[col];
                if (relu) v = fmaxf(v, 0.f);
                D[(size_t)row * ldd + col] = v;
            }
        }
    }
}

// ---------------- host-side orchestration ----------------
struct WsPtrs {
    float *L, *dinv, *Ta, *Tb, *Ha, *Hb, *acc, *Wp, *bp;
};

static void launch_pack(hipStream_t s, const float* src, float* dst,
                        int rows, int cols, int rowsP, int colsP) {
    int total = rowsP * colsP;
    pack_pad_kernel<<<(total + 255) / 256, 256, 0, s>>>(src, dst, rows, cols, colsP, total);
}

static void launch_gemm(hipStream_t s, const float* A, const float* B, const float* C,
                        float* D, const float* bias, int K, int Np,
                        int lda, int ldb, int ldc, int ldd,
                        float alpha, float beta, int relu) {
    dim3 grid((Np + 63) / 64, NNODES / 128);
    size_t shmem = 30720; // 2x(128x20) A + 2x(16x80) B floats
    wmma_gemm_f32<<<grid, 256, shmem, s>>>(A, B, C, D, bias, K, Np, lda, ldb, ldc, ldd,
                                           alpha, beta, relu);
}

static void cheb_layer(hipStream_t s, const WsPtrs& w, const float* Hin, float* Hout,
                       const float* Wl, const float* bl,
                       int k, int cin, int cinP, int cout, int coutP) {
    launch_pack(s, bl, w.bp, 1, cout, 1, coutP);
    // k=0 term: acc = T0 @ W0
    launch_pack(s, Wl, w.Wp, cin, cout, cinP, coutP);
    bool last = (k == 1);
    launch_gemm(s, Hin, w.Wp, w.acc, last ? Hout : w.acc, last ? w.bp : nullptr,
                cinP, coutP, cinP, coutP, coutP, coutP, 1.f, 0.f, last ? 1 : 0);
    if (k == 1) return;
    // T1 = L @ T0
    launch_gemm(s, w.L, Hin, nullptr, w.Ta, nullptr,
                NNODES, cinP, NNODES, cinP, cinP, cinP, 1.f, 0.f, 0);
    launch_pack(s, Wl + (size_t)cin * cout, w.Wp, cin, cout, cinP, coutP);
    last = (k == 2);
    launch_gemm(s, w.Ta, w.Wp, w.acc, last ? Hout : w.acc, last ? w.bp : nullptr,
                cinP, coutP, cinP, coutP, coutP, coutP, 1.f, 1.f, last ? 1 : 0);
    const float* Tm2 = Hin;
    const float* Tm1 = w.Ta;
    for (int i = 2; i < k; ++i) {
        // T_i = 2*(L @ T_{i-1}) - T_{i-2}; in-place over T_{i-2}'s buffer for i>2
        float* dst = (i == 2) ? w.Tb : (float*)Tm2;
        launch_gemm(s, w.L, Tm1, Tm2, dst, nullptr,
                    NNODES, cinP, NNODES, cinP, cinP, cinP, 2.f, -1.f, 0);
        launch_pack(s, Wl + (size_t)i * cin * cout, w.Wp, cin, cout, cinP, coutP);
        last = (i == k - 1);
        launch_gemm(s, dst, w.Wp, w.acc, last ? Hout : w.acc, last ? w.bp : nullptr,
                    cinP, coutP, cinP, coutP, coutP, coutP, 1.f, 1.f, last ? 1 : 0);
        Tm2 = Tm1;
        Tm1 = dst;
    }
}

extern "C" void kernel_launch(void* const* d_in, const int* in_sizes, int n_in,
                              void* d_out, int out_size, void* d_ws, size_t ws_size,
                              hipStream_t stream) {
    const float* x    = (const float*)d_in[0];
    const float* adj  = (const float*)d_in[1];
    const float* W1   = (const float*)d_in[2];
    const float* b1   = (const float*)d_in[3];
    const float* W2   = (const float*)d_in[4];
    const float* b2   = (const float*)d_in[5];
    const float* W3   = (const float*)d_in[6];
    const float* b3   = (const float*)d_in[7];
    const float* W4   = (const float*)d_in[8];
    const float* b4   = (const float*)d_in[9];
    const float* fc1w = (const float*)d_in[10];
    const float* fc1b = (const float*)d_in[11];
    const float* fc2w = (const float*)d_in[12];
    const float* fc2b = (const float*)d_in[13];
    float* out = (float*)d_out;

    float* ws = (float*)d_ws;
    const size_t NT = (size_t)NNODES * 1120; // widest feature buffer
    WsPtrs w;
    size_t off = 0;
    w.L    = ws + off; off += (size_t)NNODES * NNODES;
    w.dinv = ws + off; off += NNODES;
    w.Ta   = ws + off; off += NT;
    w.Tb   = ws + off; off += NT;
    w.Ha   = ws + off; off += NT;
    w.Hb   = ws + off; off += NT;
    w.acc  = ws + off; off += NT;
    w.Wp   = ws + off; off += 1120 * 1120;
    w.bp   = ws + off; off += 1184;

    // Laplacian
    deg_rsqrt_kernel<<<NNODES, 256, 0, stream>>>(adj, w.dinv);
    laplacian_kernel<<<dim3(NNODES / 256, NNODES), 256, 0, stream>>>(adj, w.dinv, w.L);

    // pack x [4096,7] -> Hb [4096,16]
    launch_pack(stream, x, w.Hb, NNODES, 7, NNODES, 16);

    cheb_layer(stream, w, w.Hb, w.Ha, W1, b1, 6, 7,    16,   1120, 1120);
    cheb_layer(stream, w, w.Ha, w.Hb, W2, b2, 5, 1120, 1120, 560,  560);
    cheb_layer(stream, w, w.Hb, w.Ha, W3, b3, 4, 560,  560,  280,  288);
    cheb_layer(stream, w, w.Ha, w.Hb, W4, b4, 3, 280,  288,  140,  144);

    // fc1: Ha = Hb @ fc1w + fc1b
    launch_pack(stream, fc1w, w.Wp, 140, 140, 144, 144);
    launch_pack(stream, fc1b, w.bp, 1, 140, 1, 144);
    launch_gemm(stream, w.Hb, w.Wp, nullptr, w.Ha, w.bp,
                144, 144, 144, 144, 144, 144, 1.f, 0.f, 0);
    // fc2: Hb = Ha @ fc2w + fc2b  (N padded to 16)
    launch_pack(stream, fc2w, w.Wp, 140, 4, 144, 16);
    launch_pack(stream, fc2b, w.bp, 1, 4, 1, 16);
    launch_gemm(stream, w.Ha, w.Wp, nullptr, w.Hb, w.bp,
                144, 16, 144, 16, 16, 16, 1.f, 0.f, 0);

    copy_out_kernel<<<(NNODES * 4 + 255) / 256, 256, 0, stream>>>(w.Hb, out);
}